// QueryAndGroup_50740743635355
// MI455X (gfx1250) — compile-verified
//
#include <hip/hip_runtime.h>

// ---------------------------------------------------------------------------
// QueryAndGroup (pointnet2 ball query + grouping) for MI455X / gfx1250.
//
// Kernel 1: ball query. d2(m,n) = |q|^2 - 2 q.p + |p|^2 computed with
//           V_WMMA_F32_16X16X4_F32 (16 queries x 16 points per WMMA, K=4).
//           32 points per loop iteration (2 WMMAs), register double-buffered
//           point loads to hide global_load_b96 latency, ballot-based ordered
//           first-64 selection with uniform early exit (~4.6K of 20K points
//           scanned on average since balls hold ~280 pts >> 64).
// Kernel 2: grouping gather (dominant, memory bound; features are L2-resident
//           at 82 MB vs 192 MB L2; stores coalesced along contiguous S).
// ---------------------------------------------------------------------------

typedef __attribute__((ext_vector_type(2))) float v2f;
typedef __attribute__((ext_vector_type(8))) float v8f;

#define BQ_B     8
#define BQ_N     20000
#define BQ_M     1024
#define BQ_C     128
#define BQ_S     64
#define BQ_R2    0.09f           // RADIUS^2
#define BQ_INV_R (1.0f / 0.3f)   // NORMALIZE_XYZ divisor
#define NF_ELEMS ((size_t)BQ_B * (BQ_C + 3) * BQ_M * BQ_S)

__device__ __forceinline__ unsigned ballot32(bool p) {
    return __builtin_amdgcn_ballot_w32(p);
}

// Extract this lane's query-row 16-bit hit mask from 8 ballot values.
// D layout: VGPR r holds row r (lanes 0-15) and row r+8 (lanes 16-31), so
// bal[r] low 16 bits = row r, high 16 bits = row r+8.
__device__ __forceinline__ unsigned row_mask16(const unsigned bal[8], int lane) {
    const int r3 = lane & 7;
    unsigned mm = bal[0];
    #pragma unroll
    for (int r = 1; r < 8; ++r) mm = (r3 == r) ? bal[r] : mm;
    return (lane & 8) ? (mm >> 16) : (mm & 0xFFFFu);
}

// One wave32 per (batch, 16-query tile). Scans N in 32-point chunks.
__global__ void __launch_bounds__(32)
ballquery_wmma_kernel(const float* __restrict__ xyz,
                      const float* __restrict__ new_xyz,
                      int* __restrict__ idx_out) {
    const int  lane  = threadIdx.x;
    const int  col   = lane & 15;       // query row / point column within tile
    const bool hi    = lane >= 16;      // K half selector
    const int  b     = blockIdx.y;
    const int  mtile = blockIdx.x;
    const int  m     = mtile * 16 + col;

    // Query for this lane's A-row.
    const float* q  = new_xyz + ((size_t)b * BQ_M + m) * 3;
    const float  qx = q[0], qy = q[1], qz = q[2];
    const float  q2 = qx * qx + qy * qy + qz * qz;

    // A (16x4, MxK): lanes 0-15 -> VGPR0=K0,VGPR1=K1 ; lanes 16-31 -> K2,K3.
    // Row m = (-2qx, -2qy, -2qz, |q|^2)
    v2f A;
    A.x = hi ? (-2.0f * qz) : (-2.0f * qx);
    A.y = hi ? q2           : (-2.0f * qy);

    const size_t row_base = ((size_t)b * BQ_M + (size_t)mtile * 16 + col) * BQ_S;
    const float* xb = xyz + (size_t)b * BQ_N * 3;

    int cnt   = 0;   // lanes 16-31 mirror lanes 0-15 exactly
    int first = 0;   // ref fills with first valid index (0 if none)

    // ---- software pipeline: preload chunk 0 (two 16-point tiles) ----
    float cx0, cy0, cz0, cx1, cy1, cz1;
    {
        const float* pa = xb + (size_t)col * 3;
        cx0 = pa[0]; cy0 = pa[1]; cz0 = pa[2];
        const float* pb = xb + (size_t)(16 + col) * 3;
        cx1 = pb[0]; cy1 = pb[1]; cz1 = pb[2];
    }

    for (int nbase = 0; nbase < BQ_N; nbase += 32) {
        // Issue next chunk's loads (stay in flight during WMMA/ballot work).
        const int nn    = nbase + 32;
        const int nsafe = (nn < BQ_N) ? nn : 0;   // last iter: benign reload
        float fx0, fy0, fz0, fx1, fy1, fz1;
        {
            const float* pa = xb + (size_t)(nsafe + col) * 3;
            fx0 = pa[0]; fy0 = pa[1]; fz0 = pa[2];
            const float* pb = xb + (size_t)(nsafe + 16 + col) * 3;
            fx1 = pb[0]; fy1 = pb[1]; fz1 = pb[2];
        }

        // ---- tile 0: points nbase .. nbase+15 ----
        const float p20 = cx0 * cx0 + cy0 * cy0 + cz0 * cz0;
        v2f Bv0;                      // B (4x16, KxN): col n = (px,py,pz,1)
        Bv0.x = hi ? cz0  : cx0;
        Bv0.y = hi ? 1.0f : cy0;
        v8f C0;                       // C[m][n] = |p_n|^2 (lane-indexed N)
        #pragma unroll
        for (int r = 0; r < 8; ++r) C0[r] = p20;
        v8f D0 = __builtin_amdgcn_wmma_f32_16x16x4_f32(
            false, A, false, Bv0, (short)0, C0, false, false);

        // ---- tile 1: points nbase+16 .. nbase+31 ----
        const float p21 = cx1 * cx1 + cy1 * cy1 + cz1 * cz1;
        v2f Bv1;
        Bv1.x = hi ? cz1  : cx1;
        Bv1.y = hi ? 1.0f : cy1;
        v8f C1;
        #pragma unroll
        for (int r = 0; r < 8; ++r) C1[r] = p21;
        v8f D1 = __builtin_amdgcn_wmma_f32_16x16x4_f32(
            false, A, false, Bv1, (short)0, C1, false, false);

        // Hit masks: merged 32-bit per query row, bit = offset within chunk.
        unsigned bal0[8], bal1[8];
        #pragma unroll
        for (int r = 0; r < 8; ++r) bal0[r] = ballot32(D0[r] < BQ_R2);
        #pragma unroll
        for (int r = 0; r < 8; ++r) bal1[r] = ballot32(D1[r] < BQ_R2);

        unsigned mask32 = row_mask16(bal0, lane)
                        | (row_mask16(bal1, lane) << 16);

        // Ordered append of the first NSAMPLE in-range indices.
        while (mask32 && cnt < BQ_S) {
            const int bit = __builtin_ctz(mask32);
            mask32 &= mask32 - 1u;
            const int nidx = nbase + bit;
            if (cnt == 0) first = nidx;
            if (lane < 16) idx_out[row_base + cnt] = nidx;
            ++cnt;
        }

        // Uniform early exit once all 16 rows have 64 samples.
        if (ballot32(cnt < BQ_S) == 0u) break;

        // Rotate double buffer.
        cx0 = fx0; cy0 = fy0; cz0 = fz0;
        cx1 = fx1; cy1 = fy1; cz1 = fz1;
    }

    if (lane < 16) {
        for (int s = cnt; s < BQ_S; ++s) idx_out[row_base + s] = first;
    }
}

// One 256-thread block per (b, m): 64 sample slots x 4 channel groups.
// Stores coalesced along contiguous S; gathers served from L2.
__global__ void __launch_bounds__(256)
group_kernel(const float* __restrict__ xyz,
             const float* __restrict__ new_xyz,
             const float* __restrict__ features,
             const int* __restrict__ idx,
             float* __restrict__ out) {
    const int tid = threadIdx.x;
    const int s   = tid & 63;
    const int g   = tid >> 6;
    const int m   = blockIdx.x;
    const int b   = blockIdx.y;

    const int n = idx[((size_t)b * BQ_M + m) * BQ_S + s];

    const float* pxyz = xyz     + ((size_t)b * BQ_N + n) * 3;
    const float* qxyz = new_xyz + ((size_t)b * BQ_M + m) * 3;

    for (int ch = g; ch < BQ_C + 3; ch += 4) {
        float v;
        if (ch < 3) {
            v = (pxyz[ch] - qxyz[ch]) * BQ_INV_R;           // grouped_xyz / R
        } else {
            v = features[((size_t)b * BQ_C + (ch - 3)) * BQ_N + n];
        }
        out[(((size_t)b * (BQ_C + 3) + ch) * BQ_M + m) * BQ_S + s] = v;
    }
}

extern "C" void kernel_launch(void* const* d_in, const int* in_sizes, int n_in,
                              void* d_out, int out_size, void* d_ws, size_t ws_size,
                              hipStream_t stream) {
    (void)in_sizes; (void)n_in; (void)out_size; (void)d_ws; (void)ws_size;

    const float* xyz      = (const float*)d_in[0];
    const float* new_xyz  = (const float*)d_in[1];
    // d_in[2] batch_distances, d_in[3] inds: acceleration hints only, unused.
    const float* features = (const float*)d_in[4];

    float* out     = (float*)d_out;
    int*   idx_out = (int*)d_out + NF_ELEMS;   // tuple output #2, int32 bits

    dim3 grid1(BQ_M / 16, BQ_B);
    ballquery_wmma_kernel<<<grid1, 32, 0, stream>>>(xyz, new_xyz, idx_out);

    dim3 grid2(BQ_M, BQ_B);
    group_kernel<<<grid2, 256, 0, stream>>>(xyz, new_xyz, features, idx_out, out);
}